// SparseMultiHeadAttention_17849884082434
// MI455X (gfx1250) — compile-verified
//
#include <hip/hip_runtime.h>
#include <hip/hip_bf16.h>
#include <math.h>

// ---------------- problem constants ----------------
#define NN 50000
#define EE 800000
#define DD 128
#define NH 8
#define HD 16

typedef float v2f __attribute__((ext_vector_type(2)));
typedef float v8f __attribute__((ext_vector_type(8)));

// float atomic max via signed/unsigned integer trick (valid for all finite floats, init = -inf)
__device__ __forceinline__ void atomicMaxFloat(float* addr, float val) {
    if (val >= 0.0f) {
        atomicMax((int*)addr, __float_as_int(val));
    } else {
        atomicMin((unsigned int*)addr, __float_as_uint(val));
    }
}

// ---------------- init workspace ----------------
__global__ void init_ws_kernel(float* __restrict__ agg, float* __restrict__ m,
                               float* __restrict__ denom) {
    size_t t = (size_t)blockIdx.x * blockDim.x + threadIdx.x;
    if (t < (size_t)NN * DD) agg[t] = 0.0f;
    if (t < (size_t)NN * NH) {
        m[t]     = -__builtin_inff();
        denom[t] = 0.0f;
    }
}

// ---------------- QKV projection: Out = x @ W  (WMMA f32 16x16x4) ----------------
// grid = (N/16, 3), block = 256 (8 waves). Each wave computes one 16x16 tile.
__global__ void qkv_proj_kernel(const float* __restrict__ x,
                                const float* __restrict__ Wq,
                                const float* __restrict__ Wk,
                                const float* __restrict__ Wv,
                                float* __restrict__ Q,
                                float* __restrict__ K,
                                float* __restrict__ V) {
    __shared__ float xs[16 * DD];   // 8 KB tile of x, shared by 8 waves
    const int row0  = blockIdx.x * 16;
    const int which = blockIdx.y;
    const float* __restrict__ W = (which == 0) ? Wq : (which == 1) ? Wk : Wv;
    float* __restrict__ Out     = (which == 0) ? Q  : (which == 1) ? K  : V;

    const int tid = threadIdx.x;
    for (int i = tid; i < 16 * DD; i += 256) {
        int r = i >> 7, c = i & 127;
        xs[i] = x[(size_t)(row0 + r) * DD + c];
    }
    __syncthreads();

    const int wave = tid >> 5;
    const int lane = tid & 31;
    const int lo   = lane & 15;     // M (for A) / N (for B,C)
    const int hi   = lane >> 4;     // K-half selector
    const int col0 = wave * 16;

    v8f c = {};
#pragma unroll
    for (int k = 0; k < DD; k += 4) {
        v2f a, b;
        a.x = xs[lo * DD + k + 2 * hi];
        a.y = xs[lo * DD + k + 2 * hi + 1];
        b.x = W[(size_t)(k + 2 * hi) * DD + col0 + lo];
        b.y = W[(size_t)(k + 2 * hi + 1) * DD + col0 + lo];
        c = __builtin_amdgcn_wmma_f32_16x16x4_f32(false, a, false, b,
                                                  (short)0, c, false, false);
    }
#pragma unroll
    for (int r = 0; r < 8; r++) {
        Out[(size_t)(row0 + r + 8 * hi) * DD + col0 + lo] = c[r];
    }
}

// ---------------- edge scores + segment max ----------------
// one wave per edge; grid = E/8, block = 256
__global__ void edge_scores_kernel(const int* __restrict__ ei,
                                   const float* __restrict__ ea,
                                   const float* __restrict__ We1,
                                   const float* __restrict__ be1,
                                   const float* __restrict__ We2,
                                   const float* __restrict__ be2,
                                   const float* __restrict__ Q,
                                   const float* __restrict__ K,
                                   float* __restrict__ sc,
                                   float* __restrict__ m) {
    const int e = blockIdx.x * 8 + (threadIdx.x >> 5);
    if (e >= EE) return;
    const int lane = threadIdx.x & 31;
    const int src = ei[e];
    const int tgt = ei[EE + e];

    // edge features (broadcast loads, L0/L2 cached)
    float eav[6];
#pragma unroll
    for (int i = 0; i < 6; i++) eav[i] = ea[(size_t)e * 6 + i];

    // edge MLP: h = gelu_exact(ea @ We1 + be1); bias = h @ We2
    float part[NH];
#pragma unroll
    for (int h = 0; h < NH; h++) part[h] = 0.0f;
#pragma unroll
    for (int t = 0; t < 4; t++) {
        const int j = lane + 32 * t;
        float hacc = be1[j];
#pragma unroll
        for (int i = 0; i < 6; i++) hacc += eav[i] * We1[(size_t)i * DD + j];
        const float g = 0.5f * hacc * (1.0f + erff(hacc * 0.70710678118654752f));
#pragma unroll
        for (int h = 0; h < NH; h++) part[h] += g * We2[(size_t)j * NH + h];
    }
    // full-wave reduction: every lane ends up with all 8 bias values
#pragma unroll
    for (int h = 0; h < NH; h++) {
#pragma unroll
        for (int off = 16; off >= 1; off >>= 1)
            part[h] += __shfl_xor(part[h], off, 32);
    }

    // per-head Q.K : lane group of 4 covers one head (16 dims)
    const int hh = lane >> 2;
    const int sub = lane & 3;
    const float* qrow = Q + (size_t)tgt * DD + hh * HD + sub * 4;
    const float* krow = K + (size_t)src * DD + hh * HD + sub * 4;
    float qk = 0.0f;
#pragma unroll
    for (int t = 0; t < 4; t++) qk += qrow[t] * krow[t];
    qk += __shfl_xor(qk, 1, 32);
    qk += __shfl_xor(qk, 2, 32);

    float bias = 0.0f;
#pragma unroll
    for (int h = 0; h < NH; h++) bias = (hh == h) ? part[h] : bias;

    const float s = qk * 0.25f + bias + be2[hh];   // scaling = HD^-0.5 = 0.25
    if (sub == 0) {
        sc[(size_t)e * NH + hh] = s;
        atomicMaxFloat(&m[(size_t)tgt * NH + hh], s);
    }
}

// ---------------- exp(s - m) + segment sum ----------------
__global__ void edge_exp_kernel(const int* __restrict__ ei,
                                float* __restrict__ sc,
                                const float* __restrict__ m,
                                float* __restrict__ denom) {
    size_t t = (size_t)blockIdx.x * blockDim.x + threadIdx.x;
    if (t >= (size_t)EE * NH) return;
    const int e = (int)(t >> 3);
    const int h = (int)(t & 7);
    const int tgt = ei[EE + e];
    const float v = expf(sc[t] - m[(size_t)tgt * NH + h]);
    sc[t] = v;
    atomicAdd(&denom[(size_t)tgt * NH + h], v);
}

// ---------------- normalize + weighted scatter aggregate ----------------
// one wave per edge
__global__ void edge_agg_kernel(const int* __restrict__ ei,
                                const float* __restrict__ sc,
                                const float* __restrict__ denom,
                                const float* __restrict__ V,
                                float* __restrict__ agg,
                                float* __restrict__ attn_out) {
    const int e = blockIdx.x * 8 + (threadIdx.x >> 5);
    if (e >= EE) return;
    const int lane = threadIdx.x & 31;
    const int src = ei[e];
    const int tgt = ei[EE + e];

    if (lane < NH) {
        attn_out[(size_t)e * NH + lane] =
            sc[(size_t)e * NH + lane] / denom[(size_t)tgt * NH + lane];
    }
#pragma unroll
    for (int t = 0; t < 4; t++) {
        const int d = lane + 32 * t;
        const int h = d >> 4;
        const float a = sc[(size_t)e * NH + h] / denom[(size_t)tgt * NH + h];
        atomicAdd(&agg[(size_t)tgt * DD + d], a * V[(size_t)src * DD + d]);
    }
}

// ---------------- output projection: out = agg @ Wo + bo (WMMA) ----------------
__global__ void out_proj_kernel(const float* __restrict__ agg,
                                const float* __restrict__ Wo,
                                const float* __restrict__ bo,
                                float* __restrict__ out) {
    __shared__ float xs[16 * DD];
    const int row0 = blockIdx.x * 16;
    const int tid = threadIdx.x;
    for (int i = tid; i < 16 * DD; i += 256) {
        int r = i >> 7, c = i & 127;
        xs[i] = agg[(size_t)(row0 + r) * DD + c];
    }
    __syncthreads();

    const int wave = tid >> 5;
    const int lane = tid & 31;
    const int lo = lane & 15;
    const int hi = lane >> 4;
    const int col0 = wave * 16;

    v8f c = {};
#pragma unroll
    for (int k = 0; k < DD; k += 4) {
        v2f a, b;
        a.x = xs[lo * DD + k + 2 * hi];
        a.y = xs[lo * DD + k + 2 * hi + 1];
        b.x = Wo[(size_t)(k + 2 * hi) * DD + col0 + lo];
        b.y = Wo[(size_t)(k + 2 * hi + 1) * DD + col0 + lo];
        c = __builtin_amdgcn_wmma_f32_16x16x4_f32(false, a, false, b,
                                                  (short)0, c, false, false);
    }
    const float bias = bo[col0 + lo];
#pragma unroll
    for (int r = 0; r < 8; r++) {
        out[(size_t)(row0 + r + 8 * hi) * DD + col0 + lo] = c[r] + bias;
    }
}

// ---------------- launch ----------------
extern "C" void kernel_launch(void* const* d_in, const int* in_sizes, int n_in,
                              void* d_out, int out_size, void* d_ws, size_t ws_size,
                              hipStream_t stream) {
    const float* x   = (const float*)d_in[0];
    const int*   ei  = (const int*)d_in[1];     // edge_index (2,E)
    const float* ea  = (const float*)d_in[2];
    const float* Wq  = (const float*)d_in[3];
    const float* Wk  = (const float*)d_in[4];
    const float* Wv  = (const float*)d_in[5];
    const float* We1 = (const float*)d_in[6];
    const float* be1 = (const float*)d_in[7];
    const float* We2 = (const float*)d_in[8];
    const float* be2 = (const float*)d_in[9];
    const float* Wo  = (const float*)d_in[10];
    const float* bo  = (const float*)d_in[11];

    float* out  = (float*)d_out;                  // (N, D)
    float* attn = out + (size_t)NN * DD;          // (E, H)

    // workspace layout (floats)
    float* Q     = (float*)d_ws;                  // N*D
    float* K     = Q + (size_t)NN * DD;           // N*D
    float* V     = K + (size_t)NN * DD;           // N*D
    float* agg   = V + (size_t)NN * DD;           // N*D
    float* sc    = agg + (size_t)NN * DD;         // E*H (raw scores -> exp)
    float* m     = sc + (size_t)EE * NH;          // N*H
    float* denom = m + (size_t)NN * NH;           // N*H

    // 1) init accumulators
    {
        int total = NN * DD;
        int blocks = (total + 255) / 256;
        init_ws_kernel<<<blocks, 256, 0, stream>>>(agg, m, denom);
    }
    // 2) Q/K/V projections (WMMA)
    {
        dim3 grid(NN / 16, 3);
        qkv_proj_kernel<<<grid, 256, 0, stream>>>(x, Wq, Wk, Wv, Q, K, V);
    }
    // 3) per-edge scores + segment max
    {
        int blocks = (EE + 7) / 8;
        edge_scores_kernel<<<blocks, 256, 0, stream>>>(ei, ea, We1, be1, We2, be2,
                                                       Q, K, sc, m);
    }
    // 4) exp + segment sum
    {
        size_t total = (size_t)EE * NH;
        int blocks = (int)((total + 255) / 256);
        edge_exp_kernel<<<blocks, 256, 0, stream>>>(ei, sc, m, denom);
    }
    // 5) normalize + weighted scatter-add
    {
        int blocks = (EE + 7) / 8;
        edge_agg_kernel<<<blocks, 256, 0, stream>>>(ei, sc, denom, V, agg, attn);
    }
    // 6) output projection (WMMA) + bias
    {
        out_proj_kernel<<<NN / 16, 256, 0, stream>>>(agg, Wo, bo, out);
    }
}